// DistanceLoss_36352603193422
// MI455X (gfx1250) — compile-verified
//
#include <hip/hip_runtime.h>
#include <hip/hip_bf16.h>

#define BB 8
#define CC 4
#define HH 128
#define WW 128
#define INF_F 1e9f
#define NPIX (BB * HH * WW)          // 131072 pixels (per-class handled inside)
#define NELEM (BB * CC * HH * WW)    // 524288 elements total

typedef float v2f __attribute__((ext_vector_type(2)));
typedef float v8f __attribute__((ext_vector_type(8)));

// Wave32 sum reduction via V_WMMA_F32_16X16X4_F32 with B = ones.
// A layout (ISA 7.12.2): lanes 0-15 hold rows M=0..15 with VGPR0=K0, VGPR1=K1;
// lanes 16-31 hold K2,K3. With a = (s, 0): rowsum(m) = s_m + s_{m+16}.
// D[m][n] = rowsum(m) for every n, so lane 0 holds rowsums 0..7 in its 8 D
// VGPRs and lane 16 holds rowsums 8..15. part(lane0)+part(lane16) = wave sum.
__device__ __forceinline__ float wave_wmma_partial(float s) {
  v2f a; a.x = s;    a.y = 0.0f;
  v2f b; b.x = 1.0f; b.y = 1.0f;
  v8f c = {};
  v8f d = __builtin_amdgcn_wmma_f32_16x16x4_f32(
      /*neg_a=*/false, a, /*neg_b=*/false, b,
      /*c_mod=*/(short)0, c, /*reuse_a=*/false, /*reuse_b=*/false);
  return d[0] + d[1] + d[2] + d[3] + d[4] + d[5] + d[6] + d[7];
}

// ---------------------------------------------------------------------------
// Kernel A: 1-D chamfer distance along H per (b,c,x) column; writes g^2.
// Also re-initializes the per-(b,c) max buffer (consumed by kernel B).
// ---------------------------------------------------------------------------
__global__ void edt_cols(const int* __restrict__ targets,
                         float* __restrict__ g2,
                         unsigned int* __restrict__ maxbuf) {
  if (blockIdx.x == 0 && threadIdx.x < BB * CC) maxbuf[threadIdx.x] = 0u;

  int tid = blockIdx.x * blockDim.x + threadIdx.x;   // 0 .. B*C*W-1 (exact)
  int x  = tid % WW;
  int bc = tid / WW;
  int c  = bc % CC;
  int b  = bc / CC;

  const int* tgt = targets + (b * HH) * WW + x;      // stride WW along y
  float* g = g2 + (bc * HH) * WW + x;

  float carry = INF_F;
  for (int y = 0; y < HH; ++y) {
    float f = (tgt[y * WW] == c) ? 0.0f : INF_F;
    float d = fminf(f, carry + 1.0f);
    carry = d;
    g[y * WW] = d;                                   // forward distance
  }
  carry = INF_F;
  for (int y = HH - 1; y >= 0; --y) {
    float f = (tgt[y * WW] == c) ? 0.0f : INF_F;
    float d = fminf(f, carry + 1.0f);
    carry = d;
    float gv = fminf(d, g[y * WW]);                  // min(fwd, bwd)
    g[y * WW] = gv * gv;                             // store g^2
  }
}

// ---------------------------------------------------------------------------
// Kernel B: exact squared EDT along W per (b,c,y) row, sqrt, in-place store,
// per-(b,c) running max via uint atomicMax on float bits (dist >= 0).
// One 128-thread block per row; row of g^2 staged in LDS.
// ---------------------------------------------------------------------------
__global__ void edt_rows(float* __restrict__ g2dist,
                         unsigned int* __restrict__ maxbuf) {
  __shared__ float row[WW];
  __shared__ float red[WW];
  const int bcy = blockIdx.x;          // 0 .. B*C*H-1
  const int x = threadIdx.x;           // 0 .. 127
  const int bc = bcy / HH;
  float* base = g2dist + bcy * WW;

  row[x] = base[x];
  __syncthreads();

  float m = 3.4e38f;
#pragma unroll 8
  for (int xp = 0; xp < WW; ++xp) {
    float dx = (float)(x - xp);
    m = fminf(m, fmaf(dx, dx, row[xp]));   // dx^2 exact (<=127^2), single add
  }
  float dist = sqrtf(m);
  base[x] = dist;                          // safe: all reads went through LDS

  red[x] = dist;
  __syncthreads();
  for (int s = WW / 2; s > 0; s >>= 1) {
    if (x < s) red[x] = fmaxf(red[x], red[x + s]);
    __syncthreads();
  }
  if (x == 0) atomicMax(maxbuf + bc, __float_as_uint(red[0]));
}

// ---------------------------------------------------------------------------
// Kernel C: per-pixel softmax over C=4, dist_map lookup (class pixel -> -max),
// weighted contribution, wave reduction via WMMA, one partial per block.
// ---------------------------------------------------------------------------
__global__ void loss_partial(const float* __restrict__ preds,
                             const int* __restrict__ targets,
                             const float* __restrict__ weight,
                             const float* __restrict__ dist,
                             const unsigned int* __restrict__ maxbuf,
                             float* __restrict__ blockSums) {
  __shared__ float wp[16];                           // 8 waves * 2 partials
  const int tid = threadIdx.x;
  const int idx = blockIdx.x * blockDim.x + tid;     // 0 .. B*H*W-1 (exact)

  const int w  = idx % WW;
  const int h  = (idx / WW) % HH;
  const int b  = idx / (HH * WW);
  const int t  = targets[idx];
  const int cs = HH * WW;                            // class stride

  const float* p = preds + ((b * CC) * HH + h) * WW + w;
  float p0 = p[0], p1 = p[cs], p2 = p[2 * cs], p3 = p[3 * cs];
  float mx = fmaxf(fmaxf(p0, p1), fmaxf(p2, p3));
  float e0 = expf(p0 - mx), e1 = expf(p1 - mx);
  float e2 = expf(p2 - mx), e3 = expf(p3 - mx);
  float inv = 1.0f / (e0 + e1 + e2 + e3);

  const float* db = dist + ((b * CC) * HH + h) * WW + w;
  const int mb = b * CC;
  float v0 = (t == 0) ? -__uint_as_float(maxbuf[mb + 0]) : db[0];
  float v1 = (t == 1) ? -__uint_as_float(maxbuf[mb + 1]) : db[cs];
  float v2 = (t == 2) ? -__uint_as_float(maxbuf[mb + 2]) : db[2 * cs];
  float v3 = (t == 3) ? -__uint_as_float(maxbuf[mb + 3]) : db[3 * cs];

  float s = (e0 * v0 * weight[0] + e1 * v1 * weight[1] +
             e2 * v2 * weight[2] + e3 * v3 * weight[3]) * inv;

  // All 256 threads active here -> EXEC all ones, WMMA legal.
  float part = wave_wmma_partial(s);
  int lane = tid & 31, wave = tid >> 5;
  if (lane == 0)       wp[wave * 2 + 0] = part;
  else if (lane == 16) wp[wave * 2 + 1] = part;
  __syncthreads();
  if (tid == 0) {
    float acc = 0.0f;
    for (int i = 0; i < 16; ++i) acc += wp[i];       // fixed order: deterministic
    blockSums[blockIdx.x] = acc;
  }
}

// ---------------------------------------------------------------------------
// Kernel D: single wave, fixed-order final reduction + scaling.
// ---------------------------------------------------------------------------
__global__ void loss_final(const float* __restrict__ blockSums,
                           const float* __restrict__ weight,
                           float* __restrict__ out, int nblocks) {
  __shared__ float sm[2];
  const int lane = threadIdx.x;                      // 0..31
  float s = 0.0f;
  for (int i = lane; i < nblocks; i += 32) s += blockSums[i];
  float part = wave_wmma_partial(s);
  if (lane == 0)  sm[0] = part;
  if (lane == 16) sm[1] = part;
  __syncthreads();
  if (lane == 0) {
    float sumw = weight[0] + weight[1] + weight[2] + weight[3];
    out[0] = (sm[0] + sm[1]) / (sumw * (float)NELEM);
  }
}

// ---------------------------------------------------------------------------
extern "C" void kernel_launch(void* const* d_in, const int* in_sizes, int n_in,
                              void* d_out, int out_size, void* d_ws, size_t ws_size,
                              hipStream_t stream) {
  const float* preds   = (const float*)d_in[0];      // [B,C,H,W] fp32
  const int*   targets = (const int*)d_in[1];        // [B,H,W] int32
  const float* weight  = (const float*)d_in[2];      // [C] fp32

  // Workspace layout: [0, 2MiB) g^2 / dist array; then maxbuf (32 u32, 256B-
  // aligned slot); then blockSums (512 f32). Total < 2.1 MiB.
  char* ws = (char*)d_ws;
  float*        g2        = (float*)ws;                           // NELEM f32
  unsigned int* maxbuf    = (unsigned int*)(ws + (size_t)NELEM * 4);
  float*        blockSums = (float*)(ws + (size_t)NELEM * 4 + 256);

  edt_cols<<<(BB * CC * WW) / 256, 256, 0, stream>>>(targets, g2, maxbuf);
  edt_rows<<<BB * CC * HH, WW, 0, stream>>>(g2, maxbuf);
  loss_partial<<<NPIX / 256, 256, 0, stream>>>(preds, targets, weight,
                                               g2, maxbuf, blockSums);
  loss_final<<<1, 32, 0, stream>>>(blockSums, weight, (float*)d_out, NPIX / 256);
}